// GATLayer_51032801411741
// MI455X (gfx1250) — compile-verified
//
#include <hip/hip_runtime.h>
#include <math.h>

// Problem constants (match reference)
#define B_  4
#define N_  8192
#define E_  262144
#define BN  (B_ * N_)          // 32768 rows
#define C_  256                // IN_DIM == OUT_DIM == H*D
#define H_  8
#define D_  32
#define LN_EPS 1e-5f

typedef float v2f __attribute__((ext_vector_type(2)));
typedef float v8f __attribute__((ext_vector_type(8)));

// ---------------------------------------------------------------------------
// WMMA wrapper: D = A(16x4 f32) * B(4x16 f32) + C  (exact fp32, CDNA5)
// ---------------------------------------------------------------------------
__device__ __forceinline__ v8f wmma_f32_k4(v2f a, v2f b, v8f c) {
  return __builtin_amdgcn_wmma_f32_16x16x4_f32(
      /*neg_a=*/false, a, /*neg_b=*/false, b,
      /*c_mod=*/(short)0, c, /*reuse_a=*/false, /*reuse_b=*/false);
}

// ---------------------------------------------------------------------------
// Kernel 1: zero int buffer
// ---------------------------------------------------------------------------
__global__ void k_zero_int(int* __restrict__ p, int n) {
  int i = blockIdx.x * blockDim.x + threadIdx.x;
  if (i < n) p[i] = 0;
}

// ---------------------------------------------------------------------------
// Kernel 2: fused QKV projection  Y = X*W + b  for Wq,Wk,Wv
// Block = 256 threads (8 waves). Block owns a 32-row strip of X (2 row-tiles)
// Each wave owns 6 of the 48 col-tiles; B fragments are reused across the
// 2 row-tiles -> halves W traffic per WMMA.
// ---------------------------------------------------------------------------
__global__ __launch_bounds__(256) void k_qkv(
    const float* __restrict__ X,
    const float* __restrict__ Wq, const float* __restrict__ bq,
    const float* __restrict__ Wk, const float* __restrict__ bk,
    const float* __restrict__ Wv, const float* __restrict__ bv,
    float* __restrict__ Qo, float* __restrict__ Ko, float* __restrict__ Vo) {
  __shared__ float lA[32 * 257];   // padded stride -> conflict-free column reads
  const int tid = threadIdx.x;
  const long rowbase = (long)blockIdx.x * 32;

  for (int i = tid; i < 32 * C_; i += 256) {
    int r = i >> 8, c = i & 255;
    lA[r * 257 + c] = X[(rowbase + r) * C_ + c];
  }
  __syncthreads();

  const int wave = tid >> 5, lane = tid & 31;
  const int m = lane & 15, kh = lane >> 4;   // kh selects K half (ISA A/B layout)

  v8f acc[2][6];
  const v8f vzero = {0.f, 0.f, 0.f, 0.f, 0.f, 0.f, 0.f, 0.f};
#pragma unroll
  for (int rt = 0; rt < 2; ++rt)
#pragma unroll
    for (int i = 0; i < 6; ++i) acc[rt][i] = vzero;

  const float* Wt[6];
  int ncol[6];
#pragma unroll
  for (int i = 0; i < 6; ++i) {
    int g = wave * 6 + i;
    int mat = g >> 4, ct = g & 15;
    Wt[i] = (mat == 0) ? Wq : ((mat == 1) ? Wk : Wv);
    ncol[i] = ct * 16 + m;
  }

  for (int kb = 0; kb < 64; ++kb) {
    int k0 = kb * 4 + 2 * kh;
    if ((kb & 7) == 0 && kb < 56) {          // sparse prefetch of W stream
#pragma unroll
      for (int i = 0; i < 6; ++i)
        __builtin_prefetch(&Wt[i][(size_t)(k0 + 32) * C_ + ncol[i]], 0, 0);
    }
    v2f a0 = {lA[m * 257 + k0],        lA[m * 257 + k0 + 1]};
    v2f a1 = {lA[(16 + m) * 257 + k0], lA[(16 + m) * 257 + k0 + 1]};
#pragma unroll
    for (int i = 0; i < 6; ++i) {
      v2f b = {Wt[i][(size_t)k0 * C_ + ncol[i]],
               Wt[i][(size_t)(k0 + 1) * C_ + ncol[i]]};
      acc[0][i] = wmma_f32_k4(a0, b, acc[0][i]);
      acc[1][i] = wmma_f32_k4(a1, b, acc[1][i]);
    }
  }

#pragma unroll
  for (int i = 0; i < 6; ++i) {
    int g = wave * 6 + i;
    int mat = g >> 4;
    const float* bias = (mat == 0) ? bq : ((mat == 1) ? bk : bv);
    float* outp = (mat == 0) ? Qo : ((mat == 1) ? Ko : Vo);
    int col = ncol[i];
    float bb = bias[col];
#pragma unroll
    for (int rt = 0; rt < 2; ++rt) {
#pragma unroll
      for (int r = 0; r < 8; ++r) {
        long row = rowbase + rt * 16 + r + 8 * kh;  // C/D: row = r + 8*(lane>>4)
        outp[row * C_ + col] = acc[rt][i][r] + bb;
      }
    }
  }
}

// ---------------------------------------------------------------------------
// Kernel 3: per-target-node incoming edge count
// ---------------------------------------------------------------------------
__global__ void k_count(const long long* __restrict__ ei, int* __restrict__ counts) {
  int i = blockIdx.x * blockDim.x + threadIdx.x;
  if (i >= B_ * E_) return;
  int b = i / E_, e = i - b * E_;
  int tgt = (int)ei[(size_t)b * 2 * E_ + E_ + e];
  atomicAdd(&counts[b * N_ + tgt], 1);
}

// ---------------------------------------------------------------------------
// Kernel 4: exclusive scan over counts (one block per batch, 1024 threads)
// ---------------------------------------------------------------------------
__global__ __launch_bounds__(1024) void k_scan(const int* __restrict__ counts,
                                               int* __restrict__ offs,
                                               int* __restrict__ cursor) {
  __shared__ int sums[1024];
  const int b = blockIdx.x, tid = threadIdx.x;
  const int PER = N_ / 1024;                 // 8 nodes per thread
  int base = b * N_ + tid * PER;
  int local[PER];
  int s = 0;
#pragma unroll
  for (int j = 0; j < PER; ++j) { local[j] = s; s += counts[base + j]; }
  sums[tid] = s;
  __syncthreads();
  if (tid == 0) {
    int run = 0;
    for (int t = 0; t < 1024; ++t) { int v = sums[t]; sums[t] = run; run += v; }
  }
  __syncthreads();
  int off0 = sums[tid];
#pragma unroll
  for (int j = 0; j < PER; ++j) {
    int o = off0 + local[j];
    offs[base + j] = o;
    cursor[base + j] = o;
  }
}

// ---------------------------------------------------------------------------
// Kernel 5: scatter edge ids into CSR lists
// ---------------------------------------------------------------------------
__global__ void k_scatter(const long long* __restrict__ ei,
                          int* __restrict__ cursor, int* __restrict__ elist) {
  int i = blockIdx.x * blockDim.x + threadIdx.x;
  if (i >= B_ * E_) return;
  int b = i / E_, e = i - b * E_;
  int tgt = (int)ei[(size_t)b * 2 * E_ + E_ + e];
  int pos = atomicAdd(&cursor[b * N_ + tgt], 1);
  elist[(size_t)b * E_ + pos] = e;
}

// ---------------------------------------------------------------------------
// Kernel 6: segment softmax + weighted V aggregation. One wave32 per node.
// Lanes map to head-dim (D=32). Per edge/head dot via shfl_xor tree.
// ---------------------------------------------------------------------------
__global__ __launch_bounds__(256) void k_agg(
    const float* __restrict__ Q, const float* __restrict__ K,
    const float* __restrict__ V, const long long* __restrict__ ei,
    const int* __restrict__ offs, const int* __restrict__ counts,
    const int* __restrict__ elist, float* __restrict__ AT) {
  int gw = (blockIdx.x * blockDim.x + threadIdx.x) >> 5;  // global wave = node
  int lane = threadIdx.x & 31;
  if (gw >= BN) return;
  int b = gw / N_;
  const float scale = 0.17677669529663687f;               // 1/sqrt(D)

  float q[H_], mx[H_], sm[H_], ac[H_];
  size_t qbase = (size_t)gw * C_;
#pragma unroll
  for (int h = 0; h < H_; ++h) {
    q[h] = Q[qbase + h * D_ + lane];
    mx[h] = -INFINITY; sm[h] = 0.f; ac[h] = 0.f;
  }

  int start = offs[gw];
  int cnt = counts[gw];
  const long long* srcp = ei + (size_t)b * 2 * E_;        // src row of edge_index
  const int* el = elist + (size_t)b * E_;

  for (int idx = 0; idx < cnt; ++idx) {
    int e = el[start + idx];
    int src = (int)srcp[e];
    size_t kb = ((size_t)b * N_ + src) * C_;
#pragma unroll
    for (int h = 0; h < H_; ++h) {
      float p = q[h] * K[kb + h * D_ + lane];
      p += __shfl_xor(p, 16, 32);
      p += __shfl_xor(p, 8, 32);
      p += __shfl_xor(p, 4, 32);
      p += __shfl_xor(p, 2, 32);
      p += __shfl_xor(p, 1, 32);
      float score = p * scale;
      float nm = fmaxf(mx[h], score);
      float f  = __expf(mx[h] - nm);                       // exp(-inf)=0 first edge
      float es = __expf(score - nm);
      float v  = V[kb + h * D_ + lane];
      sm[h] = sm[h] * f + es;
      ac[h] = ac[h] * f + es * v;
      mx[h] = nm;
    }
  }

#pragma unroll
  for (int h = 0; h < H_; ++h) {
    float o = (cnt > 0) ? ac[h] / fmaxf(sm[h], 1e-30f) : 0.f;
    AT[qbase + h * D_ + lane] = o;
  }
}

// ---------------------------------------------------------------------------
// Kernel 7: out-proj GEMM + bias + residual + LayerNorm, fused per 32-row strip
// 128 threads (4 waves x 4 col-tiles x 2 row-tiles).
// ---------------------------------------------------------------------------
__global__ __launch_bounds__(128) void k_out(
    const float* __restrict__ AT, const float* __restrict__ Wo,
    const float* __restrict__ bo, const float* __restrict__ X,
    const float* __restrict__ gamma, const float* __restrict__ beta,
    float* __restrict__ out) {
  __shared__ float lA[32 * 257];
  __shared__ float ps1[32][4];
  __shared__ float ps2[32][4];
  __shared__ float mu[32];
  __shared__ float rs[32];
  const int tid = threadIdx.x;
  const long rowbase = (long)blockIdx.x * 32;

  for (int i = tid; i < 32 * C_; i += 128) {
    int r = i >> 8, c = i & 255;
    lA[r * 257 + c] = AT[(rowbase + r) * C_ + c];
  }
  __syncthreads();

  const int wave = tid >> 5, lane = tid & 31;
  const int m = lane & 15, kh = lane >> 4;

  v8f acc[2][4];
  const v8f vzero = {0.f, 0.f, 0.f, 0.f, 0.f, 0.f, 0.f, 0.f};
  int ncol[4];
#pragma unroll
  for (int i = 0; i < 4; ++i) {
    acc[0][i] = vzero; acc[1][i] = vzero;
    ncol[i] = (wave * 4 + i) * 16 + m;
  }

  for (int kb = 0; kb < 64; ++kb) {
    int k0 = kb * 4 + 2 * kh;
    if ((kb & 7) == 0 && kb < 56) {
#pragma unroll
      for (int i = 0; i < 4; ++i)
        __builtin_prefetch(&Wo[(size_t)(k0 + 32) * C_ + ncol[i]], 0, 0);
    }
    v2f a0 = {lA[m * 257 + k0],        lA[m * 257 + k0 + 1]};
    v2f a1 = {lA[(16 + m) * 257 + k0], lA[(16 + m) * 257 + k0 + 1]};
#pragma unroll
    for (int i = 0; i < 4; ++i) {
      v2f b = {Wo[(size_t)k0 * C_ + ncol[i]], Wo[(size_t)(k0 + 1) * C_ + ncol[i]]};
      acc[0][i] = wmma_f32_k4(a0, b, acc[0][i]);
      acc[1][i] = wmma_f32_k4(a1, b, acc[1][i]);
    }
  }
  __syncthreads();  // everyone done reading lA; reuse as result buffer

#pragma unroll
  for (int i = 0; i < 4; ++i) {
    int col = ncol[i];
    float bb = bo[col];
#pragma unroll
    for (int rt = 0; rt < 2; ++rt) {
#pragma unroll
      for (int r = 0; r < 8; ++r) {
        int row = rt * 16 + r + 8 * kh;
        lA[row * 257 + col] = acc[rt][i][r] + bb + X[(rowbase + row) * C_ + col];
      }
    }
  }
  __syncthreads();

  {   // LayerNorm partials: 4 threads per row, 64 cols each
    int row = tid >> 2, g = tid & 3;
    float s1 = 0.f, s2 = 0.f;
    for (int c = g * 64; c < g * 64 + 64; ++c) {
      float v = lA[row * 257 + c];
      s1 += v; s2 += v * v;
    }
    ps1[row][g] = s1; ps2[row][g] = s2;
  }
  __syncthreads();
  if ((tid & 3) == 0) {
    int row = tid >> 2;
    float s1 = 0.f, s2 = 0.f;
#pragma unroll
    for (int g = 0; g < 4; ++g) { s1 += ps1[row][g]; s2 += ps2[row][g]; }
    float m_ = s1 * (1.f / 256.f);
    float var = s2 * (1.f / 256.f) - m_ * m_;
    mu[row] = m_;
    rs[row] = rsqrtf(fmaxf(var, 0.f) + LN_EPS);
  }
  __syncthreads();

  for (int i = tid; i < 32 * C_; i += 128) {
    int r = i >> 8, c = i & 255;
    float y = (lA[r * 257 + c] - mu[r]) * rs[r] * gamma[c] + beta[c];
    out[(rowbase + r) * C_ + c] = y;
  }
}

// ---------------------------------------------------------------------------
// Host launcher
// ---------------------------------------------------------------------------
extern "C" void kernel_launch(void* const* d_in, const int* in_sizes, int n_in,
                              void* d_out, int out_size, void* d_ws, size_t ws_size,
                              hipStream_t stream) {
  (void)in_sizes; (void)n_in; (void)out_size; (void)ws_size;
  const float*     X     = (const float*)d_in[0];
  const long long* ei    = (const long long*)d_in[1];   // int64 [B,2,E]
  const float*     Wq    = (const float*)d_in[2];
  const float*     bq    = (const float*)d_in[3];
  const float*     Wk    = (const float*)d_in[4];
  const float*     bk    = (const float*)d_in[5];
  const float*     Wv    = (const float*)d_in[6];
  const float*     bv    = (const float*)d_in[7];
  const float*     Wo    = (const float*)d_in[8];
  const float*     bo    = (const float*)d_in[9];
  const float*     gamma = (const float*)d_in[10];
  const float*     beta  = (const float*)d_in[11];
  float* out = (float*)d_out;

  // Workspace carve-up
  float* Qw = (float*)d_ws;
  float* Kw = Qw + (size_t)BN * C_;
  float* Vw = Kw + (size_t)BN * C_;
  float* AT = Vw + (size_t)BN * C_;
  int* counts = (int*)(AT + (size_t)BN * C_);
  int* offs   = counts + BN;
  int* cursor = offs + BN;
  int* elist  = cursor + BN;                 // B*E ints

  k_zero_int<<<(BN + 255) / 256, 256, 0, stream>>>(counts, BN);
  k_qkv<<<BN / 32, 256, 0, stream>>>(X, Wq, bq, Wk, bk, Wv, bv, Qw, Kw, Vw);
  k_count<<<(B_ * E_ + 255) / 256, 256, 0, stream>>>(ei, counts);
  k_scan<<<B_, 1024, 0, stream>>>(counts, offs, cursor);
  k_scatter<<<(B_ * E_ + 255) / 256, 256, 0, stream>>>(ei, cursor, elist);
  k_agg<<<BN / 8, 256, 0, stream>>>(Qw, Kw, Vw, ei, offs, counts, elist, AT);
  k_out<<<BN / 32, 128, 0, stream>>>(AT, Wo, bo, X, gamma, beta, out);
}